// Dconv_localshuffle_21827023798977
// MI455X (gfx1250) — compile-verified
//
#include <hip/hip_runtime.h>

#define BATCH 32
#define CIN   64
#define HH    128
#define WWID  128
#define COUT  64
#define KTOT  (CIN * 9)        // 576
#define CHALF 32
#define KHALF (CHALF * 9)      // 288
#define ROWLEN (WWID + 2)      // 130 (zero pad col 0 and 129)
#define XS_FLOATS (CHALF * 3 * ROWLEN)   // 12480 floats = 49,920 B

typedef __attribute__((ext_vector_type(2))) float v2f;
typedef __attribute__((ext_vector_type(8))) float v8f;

__global__ __launch_bounds__(256) void conv_shuffle_wmma_f32(
    const float* __restrict__ x,      // [B, C, H, W]
    const float* __restrict__ wgt,    // [OUT, C, 3, 3] -> flat [n][k], k contiguous
    const int*   __restrict__ idx,    // [C, H*W]
    float*       __restrict__ out)    // [B, OUT, H, W]
{
    __shared__ __attribute__((aligned(16))) int   koff_tab[KTOT];
    __shared__ __attribute__((aligned(16))) float xs[XS_FLOATS];

    const int tid  = threadIdx.x;
    const int lane = tid & 31;
    const int wv   = tid >> 5;        // wave id 0..7  -> M tile
    const int hi   = lane >> 4;       // half-wave selector (K offset +2)
    const int nl   = lane & 15;       // M index (A) / N index (B)

    const int blk = blockIdx.x;       // 0..B*H-1
    const int b   = blk >> 7;
    const int h   = blk & (HH - 1);

    // Warm the (L2-resident) weights with speculative prefetches.
    for (int p = tid * 16; p < KTOT * COUT; p += 256 * 16)
        __builtin_prefetch(wgt + p, 0, 1);

    // k -> LDS offset table: k = c*9 + r*3 + kw
    // koff_tab[k] = (c % 32)*3*ROWLEN + r*ROWLEN + kw
    for (int k = tid; k < KTOT; k += 256) {
        int c   = k / 9;
        int rem = k - c * 9;
        int r   = rem / 3;
        int kw  = rem - r * 3;
        koff_tab[k] = (c & (CHALF - 1)) * (3 * ROWLEN) + r * ROWLEN + kw;
    }

    v8f acc[4] = {};                  // 4 N-tiles of 16x16 fp32
    const int w0 = wv * 16;           // this wave's pixel base

    for (int half = 0; half < 2; ++half) {
        __syncthreads();
        // Stage shuffled input rows h-1..h+1 for 32 channels, zero-padded.
        const int cbase = half * CHALF;
        for (int e = tid; e < XS_FLOATS; e += 256) {
            int c   = e / (3 * ROWLEN);
            int rem = e - c * (3 * ROWLEN);
            int r   = rem / ROWLEN;
            int col = rem - r * ROWLEN;
            float v = 0.0f;
            int hh = h + r - 1;
            if (col >= 1 && col <= WWID && hh >= 0 && hh < HH) {
                int cg = cbase + c;
                int j  = idx[cg * (HH * WWID) + hh * WWID + (col - 1)];
                v = x[(b * CIN + cg) * (HH * WWID) + j];
            }
            xs[e] = v;
        }
        __syncthreads();

        const int kbase = half * KHALF;
        for (int k0 = kbase; k0 < kbase + KHALF; k0 += 4) {
            const int ka = k0 + 2 * hi;                    // even -> 8B aligned
            const int2 kofs = *reinterpret_cast<const int2*>(&koff_tab[ka]);

            // A fragment: A[m = nl][k = ka], A[m = nl][k = ka+1]
            v2f afrag;
            afrag.x = xs[kofs.x + w0 + nl];
            afrag.y = xs[kofs.y + w0 + nl];

            #pragma unroll
            for (int nt = 0; nt < 4; ++nt) {
                // B fragment: B[ka][n], B[ka+1][n]; contiguous in wgt[n][k]
                const int n = nt * 16 + nl;
                const v2f bfrag = *reinterpret_cast<const v2f*>(wgt + n * KTOT + ka);
                acc[nt] = __builtin_amdgcn_wmma_f32_16x16x4_f32(
                    /*neg_a=*/false, afrag,
                    /*neg_b=*/false, bfrag,
                    /*c_mod=*/(short)0, acc[nt],
                    /*reuse_a=*/false, /*reuse_b=*/false);
            }
        }
    }

    // Epilogue: transpose each 16x16 tile through LDS (reuse xs) so global
    // stores are contiguous in W across lanes.
    __syncthreads();
    float* tbuf = &xs[wv * 256];
    #pragma unroll 1
    for (int nt = 0; nt < 4; ++nt) {
        #pragma unroll
        for (int v = 0; v < 8; ++v) {
            // D[m][n]: m = v + 8*hi, n = nl
            tbuf[(v + 8 * hi) * 16 + nl] = acc[nt][v];
        }
        __syncthreads();
        #pragma unroll
        for (int j = 0; j < 8; ++j) {
            int ol  = hi + 2 * j;          // local out channel 0..15
            float v = tbuf[nl * 16 + ol];  // m = nl
            int o   = nt * 16 + ol;
            out[((b * COUT + o) * HH + h) * WWID + w0 + nl] = v;
        }
        __syncthreads();
    }
}

extern "C" void kernel_launch(void* const* d_in, const int* in_sizes, int n_in,
                              void* d_out, int out_size, void* d_ws, size_t ws_size,
                              hipStream_t stream) {
    const float* x   = (const float*)d_in[0];
    const float* wgt = (const float*)d_in[1];
    const int*   idx = (const int*)d_in[2];
    float*       out = (float*)d_out;

    dim3 grid(BATCH * HH);   // one block per (b, h) output row
    dim3 block(256);         // 8 waves (wave32)
    hipLaunchKernelGGL(conv_shuffle_wmma_f32, grid, block, 0, stream,
                       x, wgt, idx, out);
}